// CGConvNet_8624294330574
// MI455X (gfx1250) — compile-verified
//
#include <hip/hip_runtime.h>
#include <hip/hip_bf16.h>
#include <math.h>

// ---------------- problem constants (fixed by the reference) ----------------
#define N_NODES   50000
#define N_EDGES   800000
#define ATOM_IN   100
#define HID       64
#define BOND_IN   41
#define N_CONV    3
#define NUM_G     500

#define NODE_TILES (N_NODES / 16)   // 3125 (exact)
#define EDGE_TILES (N_EDGES / 16)   // 50000 (exact)

typedef __attribute__((ext_vector_type(2))) float v2f;
typedef __attribute__((ext_vector_type(8))) float v8f;

// fp32 WMMA: D(16x16) = A(16x4) * B(4x16) + C   -> v_wmma_f32_16x16x4_f32
__device__ __forceinline__ v8f wmma_f32x4(v2f a, v2f b, v8f c) {
  return __builtin_amdgcn_wmma_f32_16x16x4_f32(
      /*neg_a=*/false, a, /*neg_b=*/false, b,
      /*c_mod=*/(short)0, c, /*reuse_a=*/false, /*reuse_b=*/false);
}

// ------------------------------------------------------------------
// lin0: out[50000,64] = relu(n_feat[50000,100] @ W[100,64] + bias)
// one wave per 16-row tile, 4 N-tiles of 16 cols, K=100 (25 k-steps)
// ------------------------------------------------------------------
__global__ void lin0_kernel(const float* __restrict__ n_feat,
                            const float* __restrict__ w,
                            const float* __restrict__ bias,
                            float* __restrict__ out) {
  int wave = (int)((blockIdx.x * blockDim.x + threadIdx.x) >> 5);
  int lane = (int)(threadIdx.x & 31);
  if (wave >= NODE_TILES) return;
  int row0 = wave * 16;
  int m  = lane & 15;
  int hi = lane >> 4;

  v8f acc[4] = {};
  for (int k0 = 0; k0 < ATOM_IN; k0 += 4) {
    int ka = k0 + 2 * hi;
    v2f a;
    a.x = n_feat[(size_t)(row0 + m) * ATOM_IN + ka + 0];
    a.y = n_feat[(size_t)(row0 + m) * ATOM_IN + ka + 1];
#pragma unroll
    for (int nt = 0; nt < 4; ++nt) {
      int n = nt * 16 + m;
      v2f b;
      b.x = w[(size_t)(ka + 0) * HID + n];
      b.y = w[(size_t)(ka + 1) * HID + n];
      acc[nt] = wmma_f32x4(a, b, acc[nt]);
    }
  }
#pragma unroll
  for (int nt = 0; nt < 4; ++nt) {
#pragma unroll
    for (int r = 0; r < 8; ++r) {
      int row = row0 + r + 8 * hi;
      int col = nt * 16 + m;
      float v = acc[nt][r] + bias[col];
      out[(size_t)row * HID + col] = v > 0.f ? v : 0.f;
    }
  }
}

// ------------------------------------------------------------------
// per-layer node GEMMs:
//   a_buf = v @ W_src            (conv_w rows [0,64))
//   b_buf = v @ W_dst + conv_b   (conv_w rows [64,128)); bias folded here
// ------------------------------------------------------------------
__global__ void node_gemm_kernel(const float* __restrict__ v,
                                 const float* __restrict__ conv_w,
                                 const float* __restrict__ conv_b,
                                 float* __restrict__ a_buf,
                                 float* __restrict__ b_buf) {
  int wave = (int)((blockIdx.x * blockDim.x + threadIdx.x) >> 5);
  int lane = (int)(threadIdx.x & 31);
  if (wave >= NODE_TILES) return;
  int row0 = wave * 16;
  int m  = lane & 15;
  int hi = lane >> 4;

  v8f accS[4] = {};
  v8f accD[4] = {};
  for (int k0 = 0; k0 < HID; k0 += 4) {
    int ka = k0 + 2 * hi;
    v2f a;
    a.x = v[(size_t)(row0 + m) * HID + ka + 0];
    a.y = v[(size_t)(row0 + m) * HID + ka + 1];
#pragma unroll
    for (int nt = 0; nt < 4; ++nt) {
      int n = nt * 16 + m;
      v2f bs, bd;
      bs.x = conv_w[(size_t)(ka + 0) * HID + n];
      bs.y = conv_w[(size_t)(ka + 1) * HID + n];
      bd.x = conv_w[(size_t)(HID + ka + 0) * HID + n];
      bd.y = conv_w[(size_t)(HID + ka + 1) * HID + n];
      accS[nt] = wmma_f32x4(a, bs, accS[nt]);
      accD[nt] = wmma_f32x4(a, bd, accD[nt]);
    }
  }
#pragma unroll
  for (int nt = 0; nt < 4; ++nt) {
#pragma unroll
    for (int r = 0; r < 8; ++r) {
      int row = row0 + r + 8 * hi;
      int col = nt * 16 + m;
      a_buf[(size_t)row * HID + col] = accS[nt][r];
      b_buf[(size_t)row * HID + col] = accD[nt][r] + conv_b[col];
    }
  }
}

// ------------------------------------------------------------------
// edge kernel: one wave per 16-edge tile.
//   z[16,64] = gdf[16,41] @ W_e[41,64]   (WMMA, K padded 41 -> 44)
//   z += a_buf[src] + b_buf[dst]
//   msg = sigmoid(z) * leaky_relu(z, 0.01)
//   atomicAdd into vnext[dst]
// ------------------------------------------------------------------
__global__ void edge_kernel(const float* __restrict__ gdf,
                            const int*   __restrict__ src,
                            const int*   __restrict__ dst,
                            const float* __restrict__ conv_w,  // [2H+41, 64]
                            const float* __restrict__ a_buf,
                            const float* __restrict__ b_buf,
                            float* __restrict__ vnext) {
  int wave = (int)((blockIdx.x * blockDim.x + threadIdx.x) >> 5);
  int lane = (int)(threadIdx.x & 31);
  if (wave >= EDGE_TILES) return;
  int e0 = wave * 16;
  int m  = lane & 15;
  int hi = lane >> 4;

  v8f acc[4] = {};
  for (int k0 = 0; k0 < 44; k0 += 4) {
    int ka = k0 + 2 * hi;
    v2f a;
    a.x = (ka + 0 < BOND_IN) ? gdf[(size_t)(e0 + m) * BOND_IN + ka + 0] : 0.f;
    a.y = (ka + 1 < BOND_IN) ? gdf[(size_t)(e0 + m) * BOND_IN + ka + 1] : 0.f;
#pragma unroll
    for (int nt = 0; nt < 4; ++nt) {
      int n = nt * 16 + m;
      v2f b;
      b.x = (ka + 0 < BOND_IN) ? conv_w[(size_t)(2 * HID + ka + 0) * HID + n] : 0.f;
      b.y = (ka + 1 < BOND_IN) ? conv_w[(size_t)(2 * HID + ka + 1) * HID + n] : 0.f;
      acc[nt] = wmma_f32x4(a, b, acc[nt]);
    }
  }

  int se[8], de[8];
#pragma unroll
  for (int r = 0; r < 8; ++r) {
    int e = e0 + r + 8 * hi;
    se[r] = src[e];
    de[r] = dst[e];
  }

#pragma unroll
  for (int nt = 0; nt < 4; ++nt) {
#pragma unroll
    for (int r = 0; r < 8; ++r) {
      int col = nt * 16 + m;
      float z = acc[nt][r]
              + a_buf[(size_t)se[r] * HID + col]
              + b_buf[(size_t)de[r] * HID + col];
      float sg = 1.f / (1.f + __expf(-z));
      float lr = z > 0.f ? z : 0.01f * z;
      atomicAdd(&vnext[(size_t)de[r] * HID + col], sg * lr);
    }
  }
}

// ------------------------------------------------------------------
// utility / tail kernels
// ------------------------------------------------------------------
__global__ void zero_kernel(float* __restrict__ p, int n) {
  int i = (int)(blockIdx.x * blockDim.x + threadIdx.x);
  if (i < n) p[i] = 0.f;
}

__global__ void pool_kernel(const float* __restrict__ v,
                            const int*   __restrict__ gids,
                            float* __restrict__ gacc) {
  int i = (int)(blockIdx.x * blockDim.x + threadIdx.x);
  if (i >= N_NODES * HID) return;
  int node = i >> 6;
  int col  = i & 63;
  atomicAdd(&gacc[(size_t)gids[node] * HID + col], v[i]);
}

__global__ void head_kernel(const float* __restrict__ gacc,
                            const float* __restrict__ lin2_w,
                            const float* __restrict__ lin2_b,
                            float* __restrict__ out) {
  int g = (int)(blockIdx.x * blockDim.x + threadIdx.x);
  if (g >= NUM_G) return;
  float acc = 0.f;
#pragma unroll 8
  for (int c = 0; c < HID; ++c) {
    float r = gacc[(size_t)g * HID + c];
    r = r > 0.f ? r : 0.f;                 // relu(graph_rep)
    acc += r * lin2_w[c];
  }
  acc += lin2_b[0];
  out[g] = acc > 0.f ? acc : 0.f;          // relu(preds)
}

// ------------------------------------------------------------------
extern "C" void kernel_launch(void* const* d_in, const int* in_sizes, int n_in,
                              void* d_out, int out_size, void* d_ws, size_t ws_size,
                              hipStream_t stream) {
  const float* n_feat = (const float*)d_in[0];
  const float* gdf    = (const float*)d_in[1];
  const int*   src    = (const int*)  d_in[2];
  const int*   dst    = (const int*)  d_in[3];
  const int*   gids   = (const int*)  d_in[4];
  // d_in[5] = num_graphs scalar (compile-time constant NUM_G used)
  const float* lin0_w = (const float*)d_in[6];
  const float* lin0_b = (const float*)d_in[7];
  const float* conv_w = (const float*)d_in[8];
  const float* conv_b = (const float*)d_in[9];
  const float* lin2_w = (const float*)d_in[10];
  const float* lin2_b = (const float*)d_in[11];
  float* out = (float*)d_out;

  // workspace carve-up (~51.3 MB total)
  char* ws = (char*)d_ws;
  size_t off = 0;
  auto carve = [&](size_t bytes) -> float* {
    float* p = (float*)(ws + off);
    off += (bytes + 255) & ~(size_t)255;
    return p;
  };
  const size_t NODE_BYTES = (size_t)N_NODES * HID * sizeof(float);
  float* v0    = carve(NODE_BYTES);
  float* v1    = carve(NODE_BYTES);
  float* a_buf = carve(NODE_BYTES);
  float* b_buf = carve(NODE_BYTES);
  float* gacc  = carve((size_t)NUM_G * HID * sizeof(float));
  (void)ws_size; (void)in_sizes; (void)n_in; (void)out_size;

  const int BLK = 256;                       // 8 wave32 per block
  const int nodeGemmGrid = (NODE_TILES + 7) / 8;
  const int edgeGrid     = (EDGE_TILES + 7) / 8;
  const int nodeElems    = N_NODES * HID;

  // 1) lin0 + relu -> v0
  lin0_kernel<<<nodeGemmGrid, BLK, 0, stream>>>(n_feat, lin0_w, lin0_b, v0);

  // 2) three CGConv layers (ping-pong v0/v1)
  float* vcur  = v0;
  float* vnext = v1;
  for (int layer = 0; layer < N_CONV; ++layer) {
    node_gemm_kernel<<<nodeGemmGrid, BLK, 0, stream>>>(vcur, conv_w, conv_b,
                                                       a_buf, b_buf);
    zero_kernel<<<(nodeElems + BLK - 1) / BLK, BLK, 0, stream>>>(vnext, nodeElems);
    edge_kernel<<<edgeGrid, BLK, 0, stream>>>(gdf, src, dst, conv_w,
                                              a_buf, b_buf, vnext);
    float* t = vcur; vcur = vnext; vnext = t;
  }

  // 3) graph pooling + head
  const int gaccElems = NUM_G * HID;
  zero_kernel<<<(gaccElems + BLK - 1) / BLK, BLK, 0, stream>>>(gacc, gaccElems);
  pool_kernel<<<(nodeElems + BLK - 1) / BLK, BLK, 0, stream>>>(vcur, gids, gacc);
  head_kernel<<<(NUM_G + BLK - 1) / BLK, BLK, 0, stream>>>(gacc, lin2_w, lin2_b, out);
}